// SetAbstraction_27728308863300
// MI455X (gfx1250) — compile-verified
//
#include <hip/hip_runtime.h>

// Problem constants (match reference setup_inputs)
#define B_      16
#define N_      32768
#define CFEAT_  128
#define NPOINT_ 2048

#define THREADS_ 1024
#define PPT_     (N_ / THREADS_)     // 32 points per thread
#define NWAVES_  (THREADS_ / 32)     // 32 waves per workgroup
#define BIG_     1e10f

// -------------------------------------------------------------------------
// FPS kernel: one workgroup (1024 threads, 32 waves) per batch.
// (x,y) live in LDS as packed float2 (256KB of the 320KB WGP LDS),
// z and running min-distance live in registers (32+32 VGPRs/thread).
// -------------------------------------------------------------------------
__global__ __launch_bounds__(THREADS_, 1)
void fps_kernel(const float* __restrict__ xyz,   // [B,N,3]
                float* __restrict__ out_xyz,     // [B,NPOINT,3]
                int* __restrict__ fps_idx)       // [B,NPOINT]
{
    extern __shared__ unsigned char smem[];
    float2* s_xy = (float2*)smem;                          // N_ float2 = 256KB
    float*  s_bd = (float*)(smem + (size_t)N_ * sizeof(float2));
    int*    s_bi = (int*)(s_bd + NWAVES_);
    int*    s_far = (int*)(s_bi + NWAVES_);

    const int b   = blockIdx.x;
    const int tid = threadIdx.x;
    const float* xb = xyz + (size_t)b * N_ * 3;

    float pz[PPT_];
    float dmin[PPT_];

    // Coalesced load: thread tid owns points p = tid + k*1024.
#pragma unroll
    for (int k = 0; k < PPT_; ++k) {
        const int p = tid + k * THREADS_;
        const float x = xb[p * 3 + 0];
        const float y = xb[p * 3 + 1];
        const float z = xb[p * 3 + 2];
        s_xy[p] = make_float2(x, y);
        pz[k]   = z;
        dmin[k] = BIG_;
    }
    __syncthreads();

    int farthest = 0;

    for (int it = 0; it < NPOINT_; ++it) {
        // Wave-uniform index -> scalar load of centroid (L2 resident).
        const int f = __builtin_amdgcn_readfirstlane(farthest);
        const float cx = xb[f * 3 + 0];
        const float cy = xb[f * 3 + 1];
        const float cz = xb[f * 3 + 2];

        if (tid == 0) {
            fps_idx[b * NPOINT_ + it] = f;
            float* o = out_xyz + ((size_t)b * NPOINT_ + it) * 3;
            o[0] = cx; o[1] = cy; o[2] = cz;
        }

        // Fused distance-update + local argmax. Strict '>' keeps the first
        // (lowest-index) maximum within a thread, matching jnp.argmax.
        float bestd = -1.0f;
        int   besti = 0;
#pragma unroll
        for (int k = 0; k < PPT_; ++k) {
            const int p = tid + k * THREADS_;
            const float2 xy = s_xy[p];
            const float dx = xy.x - cx;
            const float dy = xy.y - cy;
            const float dz = pz[k] - cz;
            const float dd = fmaf(dx, dx, fmaf(dy, dy, dz * dz));
            const float nd = fminf(dmin[k], dd);
            dmin[k] = nd;
            const bool better = (nd > bestd);
            bestd = better ? nd : bestd;
            besti = better ? p  : besti;
        }

        // Wave32 reduction (ties -> lower index).
#pragma unroll
        for (int off = 16; off > 0; off >>= 1) {
            const float od = __shfl_xor(bestd, off, 32);
            const int   oi = __shfl_xor(besti, off, 32);
            const bool better = (od > bestd) || ((od == bestd) && (oi < besti));
            bestd = better ? od : bestd;
            besti = better ? oi : besti;
        }

        const int wave = tid >> 5;
        const int lane = tid & 31;
        if (lane == 0) { s_bd[wave] = bestd; s_bi[wave] = besti; }
        __syncthreads();

        if (wave == 0) {
            bestd = s_bd[lane];
            besti = s_bi[lane];
#pragma unroll
            for (int off = 16; off > 0; off >>= 1) {
                const float od = __shfl_xor(bestd, off, 32);
                const int   oi = __shfl_xor(besti, off, 32);
                const bool better = (od > bestd) || ((od == bestd) && (oi < besti));
                bestd = better ? od : bestd;
                besti = better ? oi : besti;
            }
            if (lane == 0) *s_far = besti;
        }
        __syncthreads();
        farthest = *s_far;
    }
}

// -------------------------------------------------------------------------
// Feature gather: one wave per selected row (128 floats = 32 lanes x b128).
// Uses the gfx1250 async global<->LDS DMA path when available.
// -------------------------------------------------------------------------
#define GATHER_BLOCK 256
#define ROWS_PER_BLOCK (GATHER_BLOCK / 32)

#if __has_builtin(__builtin_amdgcn_global_load_async_to_lds_b128) && \
    __has_builtin(__builtin_amdgcn_global_store_async_from_lds_b128)
#define USE_ASYNC_LDS 1
#else
#define USE_ASYNC_LDS 0
#endif

#if USE_ASYNC_LDS
typedef int v4i_t __attribute__((ext_vector_type(4)));
typedef __attribute__((address_space(1))) v4i_t* gv4i_t;   // global int4*
typedef __attribute__((address_space(3))) v4i_t* lv4i_t;   // LDS int4*
#endif

__global__ __launch_bounds__(GATHER_BLOCK)
void gather_kernel(const float* __restrict__ feat,      // [B,N,CFEAT]
                   const int* __restrict__ fps_idx,     // [B,NPOINT]
                   float* __restrict__ out_feat)        // [B,NPOINT,CFEAT]
{
    const int row  = blockIdx.x * ROWS_PER_BLOCK + (threadIdx.x >> 5);
    const int lane = threadIdx.x & 31;
    const int b    = row / NPOINT_;
    const int idx  = fps_idx[row];

    const float4* src = (const float4*)(feat + ((size_t)b * N_ + idx) * CFEAT_) + lane;
    float4*       dst = (float4*)(out_feat + (size_t)row * CFEAT_) + lane;

#if USE_ASYNC_LDS
    __shared__ float4 stage[ROWS_PER_BLOCK][32];
    float4* sp = &stage[threadIdx.x >> 5][lane];

    __builtin_amdgcn_global_load_async_to_lds_b128(
        (gv4i_t)(float4*)src,   // cast away const, then to AS1 int4*
        (lv4i_t)sp,
        0, 0);
#if __has_builtin(__builtin_amdgcn_s_wait_asynccnt)
    __builtin_amdgcn_s_wait_asynccnt(0);
#else
    asm volatile("s_wait_asynccnt 0x0" ::: "memory");
#endif
    __builtin_amdgcn_global_store_async_from_lds_b128(
        (gv4i_t)dst,
        (lv4i_t)sp,
        0, 0);
#if __has_builtin(__builtin_amdgcn_s_wait_asynccnt)
    __builtin_amdgcn_s_wait_asynccnt(0);
#else
    asm volatile("s_wait_asynccnt 0x0" ::: "memory");
#endif
#else
    *dst = *src;
#endif
}

// -------------------------------------------------------------------------
// Launch
// -------------------------------------------------------------------------
extern "C" void kernel_launch(void* const* d_in, const int* in_sizes, int n_in,
                              void* d_out, int out_size, void* d_ws, size_t ws_size,
                              hipStream_t stream) {
    const float* xyz  = (const float*)d_in[0];   // [B,N,3]
    const float* feat = (const float*)d_in[1];   // [B,N,CFEAT]
    // d_in[2] = npoint scalar (compile-time constant NPOINT_ here)

    float* out_xyz  = (float*)d_out;                                 // [B,NPOINT,3]
    float* out_feat = (float*)d_out + (size_t)B_ * NPOINT_ * 3;      // [B,NPOINT,CFEAT]
    int*   fps_idx  = (int*)d_ws;                                    // [B,NPOINT] scratch

    const size_t lds_bytes = (size_t)N_ * sizeof(float2)
                           + NWAVES_ * (sizeof(float) + sizeof(int))
                           + 2 * sizeof(int);

    fps_kernel<<<B_, THREADS_, lds_bytes, stream>>>(xyz, out_xyz, fps_idx);

    const int rows = B_ * NPOINT_;
    gather_kernel<<<rows / ROWS_PER_BLOCK, GATHER_BLOCK, 0, stream>>>(feat, fps_idx, out_feat);
}